// StructureEncoder_7009386627270
// MI455X (gfx1250) — compile-verified
//
#include <hip/hip_runtime.h>

typedef __attribute__((ext_vector_type(2))) float v2f;
typedef __attribute__((ext_vector_type(8))) float v8f;

#define NJ 24
#define NWAVES 8
#define NTHREADS (NWAVES * 32)
#define TILES 4
#define SPW (TILES * 16)   /* samples per wave  = 64  */
#define SPB (NWAVES * SPW) /* samples per block = 512 */

__device__ __forceinline__ v8f wmma4(v2f a, v2f b, v8f c) {
  // D = A(16x4 f32) * B(4x16 f32) + C(16x16 f32)
  return __builtin_amdgcn_wmma_f32_16x16x4_f32(false, a, false, b, (short)0, c,
                                               false, false);
}

// single-instruction in-place ReLU
__device__ __forceinline__ float relu1(float x) {
  asm("v_max_num_f32 %0, %0, 0" : "+v"(x));
  return x;
}

// cross-half lane swap via LDS-permute hardware (always called unconditionally)
__device__ __forceinline__ float lx16(float v, int pix) {
  return __int_as_float(__builtin_amdgcn_ds_bpermute(pix, __float_as_int(v)));
}

// hidden-unit permutation: C1 position m -> logical hidden (or -1 = zero pad)
// positions: [0,1,4,5,8,9,-,- | 2,3,6,7,10,11,-,-]
__device__ __forceinline__ int sigma(int m) {
  if ((m & 7) >= 6) return -1;
  return 4 * ((m & 7) >> 1) + 2 * (m >> 3) + (m & 1);
}
// feat position: C2 position m -> logical feat (or -1): [0,1,2,3,-..- | 4,5,-..]
__device__ __forceinline__ int phi(int m) {
  if (m < 4) return m;
  if (m == 8 || m == 9) return m - 4;
  return -1;
}
// child W1b K-position (k-4 = 0..5) -> logical parent feat {0,1,4,5,2,3}
__device__ __forceinline__ int rho(int x) {
  return (x < 2) ? x : ((x < 4) ? x + 2 : x - 2);
}

__global__ __launch_bounds__(NTHREADS) void skel_mlp_wmma(
    const float* __restrict__ quat, const float* __restrict__ gW1a,
    const float* __restrict__ gW1b, const float* __restrict__ gb1,
    const float* __restrict__ gW2, const float* __restrict__ gb2,
    float* __restrict__ out, int Btot) {
  // permuted+padded transposed weights: W1T[m=0..15][k=0..11], W2T likewise
  __shared__ float sW1T[NJ * 192];
  __shared__ float sW2T[NJ * 192];
  __shared__ float sB1[NJ * 16];
  __shared__ float sB2[NJ * 16];

  const int tid = threadIdx.x;
  for (int idx = tid; idx < NJ * 192; idx += NTHREADS) {
    const int i = idx / 192, rem = idx % 192, m = rem / 12, k = rem % 12;
    const int j = sigma(m);  // logical hidden for layer-1 M position
    float w1 = 0.f;
    if (j >= 0) {
      if (k < 4)
        w1 = gW1a[i * 40 + k * 10 + j];  // quat K-slots: natural order
      else if (k < 10)
        w1 = gW1b[i * 60 + rho(k - 4) * 10 + j];  // permuted feat K-slots
    }
    const int f = phi(m);  // logical feat for layer-2 M position
    float w2 = 0.f;
    if (f >= 0 && k < 10) w2 = gW2[i * 60 + k * 6 + f];  // K = logical hidden
    sW1T[idx] = w1;
    sW2T[idx] = w2;
  }
  for (int idx = tid; idx < NJ * 16; idx += NTHREADS) {
    const int i = idx >> 4, r = idx & 15;
    const int j = sigma(r), f = phi(r);
    sB1[idx] = (j >= 0) ? gb1[i * 10 + j] : 0.f;
    sB2[idx] = (f >= 0) ? gb2[i * 6 + f] : 0.f;
  }
  __syncthreads();

  const int lane = tid & 31;
  const int wv = tid >> 5;
  const int hf = lane >> 4;  // 0: lanes 0-15, 1: lanes 16-31
  const int n = lane & 15;   // sample column within a tile
  const int pix = (lane ^ 16) << 2;

  // 32-bit offsets (both arrays < 4GB) -> saddr + voffset + imm addressing.
  // OOB tiles store to sample uB-1 but load tile 0's sample (clamped to uB-1): they then
  // compute that sample's values and store them to sample uB-1's
  // address (benign only when tile 0 is OOB as well); no per-tile store guards.
  const unsigned s0 = blockIdx.x * SPB + wv * SPW + n;
  const unsigned uB = (unsigned)Btot;
  const unsigned qbase = (s0 < uB ? s0 : uB - 1) * (NJ * 16) + 8 * hf;

  const float* qpT[TILES];
  float* opT[TILES];
#pragma unroll
  for (int t = 0; t < TILES; ++t) {
    const unsigned st = s0 + 16 * t;
    const unsigned sc = st < uB ? st : uB - 1;
    // tile t's quat row sits t*6144B past tile 0's (select keeps tail in-bounds)
    const unsigned qoff = qbase + ((st < uB) ? t * (16 * NJ * 16) : 0);
    qpT[t] = (const float*)((const char*)quat + (size_t)qoff);
    opT[t] = (float*)((char*)out + (size_t)(sc * (NJ * 24)));
  }

  const float* w1t = &sW1T[n * 12 + 2 * hf];
  const float* w2t = &sW2T[n * 12 + 2 * hf];
  const float* pb1 = &sB1[8 * hf];
  const float* pb2 = &sB2[8 * hf];

  constexpr int P[NJ] = {-1, 0, 0, 0,  1,  2,  3,  4,  5,  6,  7,  8,
                         9,  9, 9, 12, 13, 14, 16, 17, 18, 19, 20, 21};
  constexpr bool HC[NJ] = {1, 1, 1, 1, 1, 1, 1, 1, 1, 1, 0, 0,
                           1, 1, 1, 0, 1, 1, 1, 1, 1, 1, 0, 0};

  // parent B-fragments = direct C2 registers (by construction of phi/rho):
  //   g1f = {f0|f4 , f1|f5} (K-step1), g2f = {f2|x , f3|x} (K-step2, x: A col=0)
  v2f g1f[TILES][NJ], g2f[TILES][NJ];

#pragma unroll
  for (int i = 0; i < NJ; ++i) {
    const int p = P[i];

    // ---- tile-invariant fragments (A matrices + bias accumulators) ----
    v2f A10, A11, A12, A20, A21, A22;
    A10.x = w1t[i * 192 + 0];
    A10.y = w1t[i * 192 + 1];
    if (p >= 0) {
      A11.x = w1t[i * 192 + 4];
      A11.y = w1t[i * 192 + 5];
      A12.x = w1t[i * 192 + 8];
      A12.y = w1t[i * 192 + 9];
    }
    A20.x = w2t[i * 192 + 0];
    A20.y = w2t[i * 192 + 1];
    A21.x = w2t[i * 192 + 4];
    A21.y = w2t[i * 192 + 5];
    A22.x = w2t[i * 192 + 8];
    A22.y = w2t[i * 192 + 9];

    v8f cb1, cb2;
#pragma unroll
    for (int r = 0; r < 8; ++r) {
      cb1[r] = pb1[i * 16 + r];
      cb2[r] = pb2[i * 16 + r];
    }

    v2f q[TILES];
#pragma unroll
    for (int t = 0; t < TILES; ++t) {
      q[t].x = qpT[t][i * 4 + 0];
      q[t].y = qpT[t][i * 4 + 1];
    }

    float res[TILES][6];
#pragma unroll
    for (int t = 0; t < TILES; ++t) {
      // ---- layer 1: h = relu(q@W1a + pf@W1b + b1), permuted columns ----
      v8f c1 = wmma4(A10, q[t], cb1);  // K 0..3 (quat)
      if (p >= 0) {
        c1 = wmma4(A11, g1f[t][p], c1);  // K 4..7 (parent C2 regs 0,1)
        c1 = wmma4(A12, g2f[t][p], c1);  // K 8..11 (parent C2 regs 2,3)
      }
#pragma unroll
      for (int r = 0; r < 6; ++r) c1[r] = relu1(c1[r]);

      // ---- layer 2: B fragments are direct C1 registers ----
      v2f B0, B1, B2;
      B0.x = c1[0];  // K {0|2}
      B0.y = c1[1];  // K {1|3}
      B1.x = c1[2];  // K {4|6}
      B1.y = c1[3];  // K {5|7}
      B2.x = c1[4];  // K {8|pad0}
      B2.y = c1[5];  // K {9|pad0}
      v8f c2 = wmma4(A20, B0, cb2);
      c2 = wmma4(A21, B1, c2);
      c2 = wmma4(A22, B2, c2);
#pragma unroll
      for (int r = 0; r < 4; ++r) c2[r] = relu1(c2[r]);

      if (HC[i]) {  // child fragments = these registers, verbatim
        g1f[t][i].x = c2[0];
        g1f[t][i].y = c2[1];
        g2f[t][i].x = c2[2];
        g2f[t][i].y = c2[3];
      }
      // logical feats: f0..f3 = regs 0..3 (lo), f4,f5 = hi half of regs 0,1
      res[t][0] = c2[0];
      res[t][1] = c2[1];
      res[t][2] = c2[2];
      res[t][3] = c2[3];
      res[t][4] = lx16(c2[0], pix);
      res[t][5] = lx16(c2[1], pix);
    }

    if (hf == 0) {  // single exec region per joint; all tiles' stores inside
#pragma unroll
      for (int t = 0; t < TILES; ++t) {
#pragma unroll
        for (int r = 0; r < 6; ++r) opT[t][i * 6 + r] = res[t][r];
      }
    }
  }
}

extern "C" void kernel_launch(void* const* d_in, const int* in_sizes, int n_in,
                              void* d_out, int out_size, void* d_ws,
                              size_t ws_size, hipStream_t stream) {
  const float* quat = (const float*)d_in[0];
  const float* W1a = (const float*)d_in[1];
  const float* W1b = (const float*)d_in[2];
  const float* b1 = (const float*)d_in[3];
  const float* W2 = (const float*)d_in[4];
  const float* b2 = (const float*)d_in[5];
  float* out = (float*)d_out;
  const int Btot = in_sizes[0] / (NJ * 4);  // quat is (B, 24, 4)
  const int blocks = (Btot + SPB - 1) / SPB;
  skel_mlp_wmma<<<blocks, NTHREADS, 0, stream>>>(quat, W1a, W1b, b1, W2, b2,
                                                 out, Btot);
}